// SpatialBranch_4131758538844
// MI455X (gfx1250) — compile-verified
//
#include <hip/hip_runtime.h>
#include <hip/hip_bf16.h>

typedef __attribute__((ext_vector_type(2))) float v2f;
typedef __attribute__((ext_vector_type(8))) float v8f;
typedef __attribute__((ext_vector_type(4))) int   v4i;

#define NN 1024
#define TT 24
#define FF 8
#define HH 64
#define PP 24
#define TOPK 16

// ---- CDNA5 async-to-LDS path (ASYNCcnt) -----------------------------------
#if defined(__has_builtin)
#if __has_builtin(__builtin_amdgcn_global_load_async_to_lds_b128) && \
    __has_builtin(__builtin_amdgcn_s_wait_asynccnt)
#define HAVE_ASYNC 1
#endif
#endif

#define GAS1 __attribute__((address_space(1)))
#define LAS3 __attribute__((address_space(3)))

#ifdef HAVE_ASYNC
#define ASYNC_CP16(ldst, gsrc)                                            \
  __builtin_amdgcn_global_load_async_to_lds_b128(                         \
      (GAS1 v4i*)(gsrc), (LAS3 v4i*)(ldst), 0, 0)
#define ASYNC_WAIT(n) __builtin_amdgcn_s_wait_asynccnt(n)
#else
#warning "gfx1250 async-to-LDS builtins not found; falling back to synchronous staging"
#define ASYNC_CP16(ldst, gsrc)                                            \
  do { *(float4*)(ldst) = *(const float4*)(gsrc); } while (0)
#define ASYNC_WAIT(n) do { } while (0)
#endif

// ---------------------------------------------------------------------------
// Kernel 1: static adjacency with exact tie-aware top-k threshold per row.
// ---------------------------------------------------------------------------
__global__ void __launch_bounds__(256) static_topk(const float* __restrict__ coords,
                                                   float* __restrict__ sadj) {
  const int i = blockIdx.x, tid = threadIdx.x;
  __shared__ float vals[NN];
  __shared__ float redf[256];
  __shared__ int   redi[256];
  __shared__ float s_bound;
  __shared__ int   s_rem;

  const float cx = coords[2 * i], cy = coords[2 * i + 1];
  for (int j = tid; j < NN; j += 256) {
    float dx = coords[2 * j] - cx;
    float dy = coords[2 * j + 1] - cy;
    vals[j] = __expf(-50.0f * (dx * dx + dy * dy));  // sigma=0.1 -> 1/(2s^2)=50
  }
  if (tid == 0) { s_bound = 3.4e38f; s_rem = TOPK; }
  __syncthreads();

  for (int iter = 0; iter < TOPK; ++iter) {
    float bound = s_bound;
    int rem = s_rem;
    __syncthreads();
    if (rem <= 0) break;
    float m = -1.0f;
    for (int j = tid; j < NN; j += 256) {
      float v = vals[j];
      if (v < bound && v > m) m = v;
    }
    redf[tid] = m;
    __syncthreads();
    for (int s = 128; s > 0; s >>= 1) {
      if (tid < s) redf[tid] = fmaxf(redf[tid], redf[tid + s]);
      __syncthreads();
    }
    m = redf[0];
    __syncthreads();
    int c = 0;
    for (int j = tid; j < NN; j += 256)
      if (vals[j] == m) ++c;
    redi[tid] = c;
    __syncthreads();
    for (int s = 128; s > 0; s >>= 1) {
      if (tid < s) redi[tid] += redi[tid + s];
      __syncthreads();
    }
    if (tid == 0) { s_bound = m; s_rem = rem - redi[0]; }
    __syncthreads();
  }
  __syncthreads();
  const float kth = s_bound;
  for (int j = tid; j < NN; j += 256) {
    float v = vals[j];
    sadj[i * NN + j] = (v >= kth) ? v : 0.0f;
  }
}

// ---------------------------------------------------------------------------
// Kernel 2: dynamic (unnormalized) adjacency at t = T-1 only, plus dinv.
// cos(wd - atan2(dy,dx)) == (cos(wd)*dx + sin(wd)*dy) * rsqrt(d2)
// ---------------------------------------------------------------------------
__global__ void __launch_bounds__(256) dyn_adj(const float* __restrict__ coords,
                                               const float* __restrict__ sadj,
                                               const float* __restrict__ ws_seq,
                                               const float* __restrict__ wd_seq,
                                               float* __restrict__ adj_un,
                                               float* __restrict__ dinv) {
  const int i = blockIdx.x, b = blockIdx.y, tid = threadIdx.x;
  __shared__ float red[256];
  const int base = (b * TT + (TT - 1)) * NN;
  const float wsv = ws_seq[base + i];
  const float wdv = wd_seq[base + i];
  const float c = __cosf(wdv), s = __sinf(wdv);
  const float cx = coords[2 * i], cy = coords[2 * i + 1];
  float* row = adj_un + ((size_t)(b * NN + i)) * NN;
  float sum = 0.0f;
  for (int j = tid; j < NN; j += 256) {
    float sa = sadj[i * NN + j];
    float val = 0.0f;
    if (sa > 0.0f) {
      float dx = coords[2 * j] - cx;
      float dy = coords[2 * j + 1] - cy;
      float d2 = dx * dx + dy * dy;
      float align = (d2 > 0.0f) ? (c * dx + s * dy) * rsqrtf(d2) : c;
      val = sa * __expf(wsv * align);
    }
    row[j] = val;
    sum += val;
  }
  red[tid] = sum;
  __syncthreads();
  for (int s2 = 128; s2 > 0; s2 >>= 1) {
    if (tid < s2) red[tid] += red[tid + s2];
    __syncthreads();
  }
  if (tid == 0) dinv[b * NN + i] = rsqrtf(red[0] + 1e-8f);
}

// ---------------------------------------------------------------------------
// Kernel 3: xs[b,j,0:16] = dinv[b,j] * x[b,T-1,j,0:8], zero-padded to 16 cols.
// ---------------------------------------------------------------------------
__global__ void __launch_bounds__(256) scale_x(const float* __restrict__ x,
                                               const float* __restrict__ dinv,
                                               float* __restrict__ xs) {
  int g = blockIdx.x * 256 + threadIdx.x;
  if (g >= 4 * NN) return;
  int b = g >> 10, j = g & (NN - 1);
  float dv = dinv[g];
  const float* xp = x + ((size_t)((b * TT + (TT - 1)) * NN + j)) * FF;
  float* op = xs + (size_t)g * 16;
#pragma unroll
  for (int f = 0; f < FF; ++f) op[f] = dv * xp[f];
#pragma unroll
  for (int f = FF; f < 16; ++f) op[f] = 0.0f;
}

// ---------------------------------------------------------------------------
// Kernel 4: first propagation + first dense layer (one wave / 16-row tile).
// ---------------------------------------------------------------------------
__global__ void __launch_bounds__(32) prop1(const float* __restrict__ adj_un,
                                            const float* __restrict__ xs,
                                            const float* __restrict__ dinv,
                                            const float* __restrict__ W1,
                                            const float* __restrict__ b1,
                                            float* __restrict__ zs) {
  const int tile = blockIdx.x;
  const int b    = blockIdx.y;
  const int row0 = tile * 16;
  const int lane = threadIdx.x & 31;
  const int half = lane >> 4, lm = lane & 15;

  const float* A  = adj_un + (size_t)b * NN * NN;
  const float* Bx = xs + (size_t)b * NN * 16;

  v8f acc = {};
  for (int k0 = 0; k0 < NN; k0 += 4) {
    v2f a = *(const v2f*)(A + (size_t)(row0 + lm) * NN + k0 + 2 * half);
    v2f bb;
    bb[0] = Bx[(k0 + 2 * half) * 16 + lm];
    bb[1] = Bx[(k0 + 2 * half + 1) * 16 + lm];
    acc = __builtin_amdgcn_wmma_f32_16x16x4_f32(false, a, false, bb,
                                                (short)0, acc, false, false);
  }

  __shared__ float ht[16][16];
#pragma unroll
  for (int v = 0; v < 8; ++v) {
    int r = v + 8 * half;
    ht[r][lm] = acc[v] * dinv[b * NN + row0 + r];
  }
  __syncthreads();

  for (int o = lane; o < 16 * HH; o += 32) {
    int r = o >> 6, hc = o & (HH - 1);
    float z = b1[hc];
#pragma unroll
    for (int f = 0; f < FF; ++f) z = fmaf(ht[r][f], W1[f * HH + hc], z);
    z = fmaxf(z, 0.0f);
    zs[((size_t)(b * NN + row0 + r)) * HH + hc] = dinv[b * NN + row0 + r] * z;
  }
}

// ---------------------------------------------------------------------------
// Kernel 5: second propagation (LDS-staged, async double-buffered) + dense
// layer + prediction head, fused. 4 waves/block; each wave owns a 16x16
// col-tile; A/B K-panels are filled with global_load_async_to_lds_b128 and
// consumed from LDS (ds_load) while the next panel streams in (ASYNCcnt).
// ---------------------------------------------------------------------------
#define KP 64          // K-panel depth
#define NPAN (NN / KP) // 16 panels
#define BPAD 68        // padded row stride (floats): 16B-aligned, bank-spread

__global__ void __launch_bounds__(128) prop2(const float* __restrict__ adj_un,
                                             const float* __restrict__ zs,
                                             const float* __restrict__ dinv,
                                             const float* __restrict__ W2,
                                             const float* __restrict__ b2,
                                             const float* __restrict__ Wp,
                                             const float* __restrict__ bp,
                                             float* __restrict__ out) {
  const int tile = blockIdx.x;
  const int b    = blockIdx.y;
  const int row0 = tile * 16;
  const int tid  = threadIdx.x;
  const int wave = tid >> 5, lane = tid & 31;
  const int half = lane >> 4, lm = lane & 15;
  const int col0 = wave * 16;

  const float* A  = adj_un + (size_t)b * NN * NN;
  const float* Bz = zs + (size_t)b * NN * HH;

  __shared__ float As[2][16][BPAD];   // A panel: 16 rows x 64 K
  __shared__ float Bs[2][KP][BPAD];   // B panel: 64 K x 64 cols
  __shared__ float h2s[16][HH];
  __shared__ float z2s[16][HH];

  // Issue one panel's async fills: A = 256 x 16B (2/thread), B = 1024 x 16B
  // (8/thread) -> 10 outstanding ASYNCcnt ops per thread per panel.
  auto fill = [&](int k0, int buf) {
#pragma unroll
    for (int i = 0; i < 2; ++i) {
      int c = tid + 128 * i;            // 0..255
      int r = c >> 4, sg = c & 15;
      ASYNC_CP16(&As[buf][r][sg * 4], A + (size_t)(row0 + r) * NN + k0 + sg * 4);
    }
#pragma unroll
    for (int i = 0; i < 8; ++i) {
      int c = tid + 128 * i;            // 0..1023
      int r = c >> 4, sg = c & 15;
      ASYNC_CP16(&Bs[buf][r][sg * 4], Bz + (size_t)(k0 + r) * HH + sg * 4);
    }
  };

  v8f acc = {};
  fill(0, 0);
  for (int p = 0; p < NPAN - 1; ++p) {
    fill((p + 1) * KP, (p + 1) & 1);
    ASYNC_WAIT(10);                 // panel p's fills (this wave) complete
    __syncthreads();                // visible block-wide
    const int buf = p & 1;
#pragma unroll
    for (int kp = 0; kp < KP; kp += 4) {
      v2f a = *(const v2f*)&As[buf][lm][kp + 2 * half];
      v2f bb;
      bb[0] = Bs[buf][kp + 2 * half][col0 + lm];
      bb[1] = Bs[buf][kp + 2 * half + 1][col0 + lm];
      acc = __builtin_amdgcn_wmma_f32_16x16x4_f32(false, a, false, bb,
                                                  (short)0, acc, false, false);
    }
    __syncthreads();                // done reading buf before it refills
  }
  ASYNC_WAIT(0);
  __syncthreads();
  {
    const int buf = (NPAN - 1) & 1;
#pragma unroll
    for (int kp = 0; kp < KP; kp += 4) {
      v2f a = *(const v2f*)&As[buf][lm][kp + 2 * half];
      v2f bb;
      bb[0] = Bs[buf][kp + 2 * half][col0 + lm];
      bb[1] = Bs[buf][kp + 2 * half + 1][col0 + lm];
      acc = __builtin_amdgcn_wmma_f32_16x16x4_f32(false, a, false, bb,
                                                  (short)0, acc, false, false);
    }
  }

#pragma unroll
  for (int v = 0; v < 8; ++v) {
    int r = v + 8 * half;
    h2s[r][col0 + lm] = acc[v] * dinv[b * NN + row0 + r];
  }
  __syncthreads();

  for (int o = tid; o < 16 * HH; o += 128) {
    int r = o >> 6, hc = o & (HH - 1);
    float z = b2[hc];
#pragma unroll 8
    for (int k = 0; k < HH; ++k) z = fmaf(h2s[r][k], W2[k * HH + hc], z);
    z2s[r][hc] = fmaxf(z, 0.0f);
  }
  __syncthreads();

  for (int o = tid; o < 16 * PP; o += 128) {
    int r = o / PP, p = o - r * PP;
    float z = bp[p];
#pragma unroll 8
    for (int k = 0; k < HH; ++k) z = fmaf(z2s[r][k], Wp[k * PP + p], z);
    out[((size_t)(b * PP + p)) * NN + row0 + r] = z;
  }
}

// ---------------------------------------------------------------------------
extern "C" void kernel_launch(void* const* d_in, const int* in_sizes, int n_in,
                              void* d_out, int out_size, void* d_ws, size_t ws_size,
                              hipStream_t stream) {
  const float* x      = (const float*)d_in[0];
  const float* ws_seq = (const float*)d_in[1];
  const float* wd_seq = (const float*)d_in[2];
  const float* coords = (const float*)d_in[3];
  const float* W1     = (const float*)d_in[4];
  const float* b1     = (const float*)d_in[5];
  const float* W2     = (const float*)d_in[6];
  const float* b2     = (const float*)d_in[7];
  const float* Wp     = (const float*)d_in[8];
  const float* bp     = (const float*)d_in[9];
  float* out = (float*)d_out;

  float* ws = (float*)d_ws;
  float* sadj   = ws;                               // N*N
  float* adj_un = sadj + (size_t)NN * NN;           // B*N*N
  float* dinv   = adj_un + (size_t)4 * NN * NN;     // B*N
  float* xs     = dinv + (size_t)4 * NN;            // B*N*16
  float* zsbuf  = xs + (size_t)4 * NN * 16;         // B*N*H

  static_topk<<<NN, 256, 0, stream>>>(coords, sadj);
  dyn_adj<<<dim3(NN, 4), 256, 0, stream>>>(coords, sadj, ws_seq, wd_seq, adj_un, dinv);
  scale_x<<<(4 * NN + 255) / 256, 256, 0, stream>>>(x, dinv, xs);
  prop1<<<dim3(NN / 16, 4), 32, 0, stream>>>(adj_un, xs, dinv, W1, b1, zsbuf);
  prop2<<<dim3(NN / 16, 4), 128, 0, stream>>>(adj_un, zsbuf, dinv, W2, b2, Wp, bp, out);
}